// AttentionDecoder_39865886441507
// MI455X (gfx1250) — compile-verified
//
#include <hip/hip_runtime.h>
#include <math.h>

#define T      10
#define E      64
#define H      64
#define DY     5
#define G4H    256
#define MB     16     // batch rows per block (one WMMA M-tile)
#define THREADS 256
#define NWAVE  8

typedef __attribute__((ext_vector_type(2))) float v2f;
typedef __attribute__((ext_vector_type(8))) float v8f;

#if __has_builtin(__builtin_amdgcn_tanhf)
#define TANHF(x) __builtin_amdgcn_tanhf(x)
#else
#define TANHF(x) tanhf(x)
#endif

#if __has_builtin(__builtin_amdgcn_rcpf)
#define RCPF(x) __builtin_amdgcn_rcpf(x)
#else
#define RCPF(x) (1.0f / (x))
#endif

__device__ __forceinline__ float sigmoidf_fast(float x) {
  return RCPF(1.0f + __expf(-x));
}

// init kernel: WhhT[k*256 + j] = W_hh[j*64 + k]  (B-operand layout, 64 KB, L2/WGP$-hot)
__global__ void transpose_whh(const float* __restrict__ W_hh, float* __restrict__ WhhT) {
  const int i = blockIdx.x * blockDim.x + threadIdx.x;
  if (i < 64 * G4H) {
    const int k = i >> 8, j = i & 255;
    WhhT[i] = W_hh[j * 64 + k];
  }
}

template <bool WHH_LDS>
__global__ __launch_bounds__(THREADS) void attn_decoder_kernel(
    const float* __restrict__ x_encoded, const float* __restrict__ y_history,
    const float* __restrict__ ht0,       const float* __restrict__ ct0,
    const float* __restrict__ attn_w1,   const float* __restrict__ attn_b1,
    const float* __restrict__ attn_w2,
    const float* __restrict__ W_ih,      const float* __restrict__ W_hh,
    const float* __restrict__ b_ih,      const float* __restrict__ b_hh,
    const float* __restrict__ fc_w,      const float* __restrict__ fc_b,
    const float* __restrict__ fco_w,     const float* __restrict__ fco_b,
    const float* __restrict__ WhhT,      // pre-transposed W_hh in d_ws (used when !WHH_LDS)
    float* __restrict__ out)
{
  // ---- LDS: ~158 KB when !WHH_LDS -> 2 blocks/WGP; ~222 KB otherwise ----
  __shared__ float sW1hc[128 * 64];          // B-operand: [k][e] = attn_w1[e][k], k<128
  __shared__ float sWhh [WHH_LDS ? 64 * G4H : 8];
  __shared__ float sX   [MB * T * E];        // x tile
  __shared__ float sEnc [MB * T * E];        // enc_proj tile
  __shared__ float sQ   [MB * E];
  __shared__ float sH   [MB * H];
  __shared__ float sC   [MB * H];
  __shared__ float sGates[MB * G4H];         // 16 KB; doubles as W1enc [d][e] in phase 1
  __shared__ float sCtx [MB * E];
  __shared__ float sYt  [MB * 8];
  __shared__ float sY   [MB * T * DY];
  __shared__ float sB1  [E];
  __shared__ float sW2  [E];
  __shared__ float sBg  [G4H];
  __shared__ float sWihT[DY * G4H];          // [d][j] = W_ih[j][d]
  __shared__ float sFcW [DY * 72];           // fc_w padded rows
  __shared__ float sFcoW[DY * 128];
  __shared__ float sFcB [8];
  __shared__ float sFcoB[8];

  const int tid = threadIdx.x;
  const int b0  = blockIdx.x * MB;
  const int wid  = tid >> 5;
  const int lane = tid & 31;
  const int l    = lane & 15;   // column (B/C/D) or row (A)
  const int hi   = lane >> 4;   // K-half selector

  // ---------------- Phase 0: cooperative staging ----------------
  for (int i = tid; i < 128 * 64; i += THREADS) { int k = i >> 6, e = i & 63; sW1hc[i] = attn_w1[e * 192 + k]; }
  for (int i = tid; i < 64 * 64;  i += THREADS) { int d = i >> 6, e = i & 63; sGates[i] = attn_w1[e * 192 + 128 + d]; }
  if constexpr (WHH_LDS) {
    for (int i = tid; i < 64 * G4H; i += THREADS) { int k = i >> 8, j = i & 255; sWhh[i] = W_hh[j * 64 + k]; }
  }
  for (int i = tid; i < DY * G4H; i += THREADS) { int d = i >> 8, j = i & 255; sWihT[i] = W_ih[j * DY + d]; }
  for (int i = tid; i < MB * T * E;  i += THREADS) sX[i] = x_encoded[(size_t)b0 * T * E + i];
  for (int i = tid; i < MB * T * DY; i += THREADS) sY[i] = y_history[(size_t)b0 * T * DY + i];
  for (int i = tid; i < MB * H; i += THREADS) { sH[i] = ht0[(size_t)b0 * H + i]; sC[i] = ct0[(size_t)b0 * H + i]; }
  if (tid < E) { sB1[tid] = attn_b1[tid]; sW2[tid] = attn_w2[tid]; }
  for (int i = tid; i < G4H; i += THREADS) sBg[i] = b_ih[i] + b_hh[i];
  for (int i = tid; i < DY * 69; i += THREADS) { int d = i / 69, j = i % 69; sFcW[d * 72 + j] = fc_w[i]; }
  for (int i = tid; i < DY * 128; i += THREADS) sFcoW[i] = fco_w[i];
  if (tid < DY) { sFcB[tid] = fc_b[tid]; sFcoB[tid] = fco_b[tid]; }
  __syncthreads();

  // ---------------- Phase 1: enc_proj = x @ W1_enc^T (WMMA fp32) ----------------
  for (int job = wid; job < T * 4; job += NWAVE) {
    const int t  = job >> 2;
    const int n0 = (job & 3) << 4;
    v8f acc = {};
#pragma unroll
    for (int ks = 0; ks < 16; ++ks) {
      const int kk = ks * 4 + hi * 2;
      v2f a, b;
      a.x = sX[l * (T * E) + t * E + kk];
      a.y = sX[l * (T * E) + t * E + kk + 1];
      b.x = sGates[kk * 64 + n0 + l];          // W1enc B-operand
      b.y = sGates[(kk + 1) * 64 + n0 + l];
      acc = __builtin_amdgcn_wmma_f32_16x16x4_f32(false, a, false, b, (short)0, acc, false, false);
    }
#pragma unroll
    for (int v = 0; v < 8; ++v)
      sEnc[(v + hi * 8) * (T * E) + t * E + n0 + l] = acc[v];
  }
  __syncthreads();

  // B-fragment source for the gates GEMM (LDS or global WGP$/L2-hot stream)
  auto whh_at = [&](int idx) -> float {
    if constexpr (WHH_LDS) return sWhh[idx];
    else                   return WhhT[idx];
  };

  // single gate tile: rawgates tile = ht @ W_hh^T + (b_ih + b_hh)
  auto gate_tile = [&](int g, const v2f* ah) {
    const int n0 = g << 4;
    v8f acc = {};
#pragma unroll
    for (int ks = 0; ks < 16; ++ks) {
      const int kk = ks * 4 + hi * 2;
      v2f b;
      b.x = whh_at(kk * G4H + n0 + l);
      b.y = whh_at((kk + 1) * G4H + n0 + l);
      acc = __builtin_amdgcn_wmma_f32_16x16x4_f32(false, ah[ks], false, b, (short)0, acc, false, false);
    }
    const int n = n0 + l;
    const float bg = sBg[n];
#pragma unroll
    for (int v = 0; v < 8; ++v)
      sGates[(v + hi * 8) * G4H + n] = acc[v] + bg;
  };

  // two gate tiles fused: independent WMMA chains + 2x memory-level parallelism
  auto gate_tile2 = [&](int g0, int g1, const v2f* ah) {
    const int n0 = g0 << 4, n1 = g1 << 4;
    v8f acc0 = {}, acc1 = {};
#pragma unroll
    for (int ks = 0; ks < 16; ++ks) {
      const int kk = ks * 4 + hi * 2;
      v2f b0, b1;
      b0.x = whh_at(kk * G4H + n0 + l);
      b0.y = whh_at((kk + 1) * G4H + n0 + l);
      b1.x = whh_at(kk * G4H + n1 + l);
      b1.y = whh_at((kk + 1) * G4H + n1 + l);
      acc0 = __builtin_amdgcn_wmma_f32_16x16x4_f32(false, ah[ks], false, b0, (short)0, acc0, false, false);
      acc1 = __builtin_amdgcn_wmma_f32_16x16x4_f32(false, ah[ks], false, b1, (short)0, acc1, false, false);
    }
    const float bg0 = sBg[n0 + l], bg1 = sBg[n1 + l];
#pragma unroll
    for (int v = 0; v < 8; ++v) {
      sGates[(v + hi * 8) * G4H + n0 + l] = acc0[v] + bg0;
      sGates[(v + hi * 8) * G4H + n1 + l] = acc1[v] + bg1;
    }
  };

  // ---------------- Phase 2: recurrence over T steps ----------------
  for (int ts = 0; ts < T; ++ts) {
    // ---- Phase A (all 8 waves, WMMA): q = [h,c]@W1hc^T + b1 ; rawgates = h@Whh^T + bg ----
    {
      v2f ah[16];
#pragma unroll
      for (int ks = 0; ks < 16; ++ks) {
        const int kk = ks * 4 + hi * 2;
        ah[ks].x = sH[l * H + kk];
        ah[ks].y = sH[l * H + kk + 1];
      }
      if (wid < 4) {
        // q n-tile = wid (K = 128 over [h, c])
        const int n0 = wid << 4;
        v8f acc = {};
#pragma unroll
        for (int ks = 0; ks < 16; ++ks) {
          const int kk = ks * 4 + hi * 2;
          v2f b;
          b.x = sW1hc[kk * 64 + n0 + l];
          b.y = sW1hc[(kk + 1) * 64 + n0 + l];
          acc = __builtin_amdgcn_wmma_f32_16x16x4_f32(false, ah[ks], false, b, (short)0, acc, false, false);
        }
#pragma unroll
        for (int ks = 0; ks < 16; ++ks) {
          const int kk = ks * 4 + hi * 2;
          v2f a, b;
          a.x = sC[l * H + kk];
          a.y = sC[l * H + kk + 1];
          b.x = sW1hc[(64 + kk) * 64 + n0 + l];
          b.y = sW1hc[(64 + kk + 1) * 64 + n0 + l];
          acc = __builtin_amdgcn_wmma_f32_16x16x4_f32(false, a, false, b, (short)0, acc, false, false);
        }
        const float bias = sB1[n0 + l];
#pragma unroll
        for (int v = 0; v < 8; ++v)
          sQ[(v + hi * 8) * E + n0 + l] = acc[v] + bias;
        gate_tile(wid, ah);                 // 1 gate tile
      } else {
        const int g0 = 4 + (wid - 4) * 3;   // 3 gate tiles (2 fused + 1)
        gate_tile2(g0 + 0, g0 + 1, ah);
        gate_tile(g0 + 2, ah);
      }
    }
    __syncthreads();

    // ---- Phase B: scores -> softmax -> context -> y_tilde. thread = (b, i) ----
    {
      const int b = tid >> 4, i = tid & 15;
      float p[T];
#pragma unroll
      for (int t = 0; t < T; ++t) p[t] = 0.f;
#pragma unroll
      for (int e4 = 0; e4 < 4; ++e4) {
        const int e = i + e4 * 16;
        const float qv = sQ[b * E + e];
        const float w  = sW2[e];
#pragma unroll
        for (int t = 0; t < T; ++t)
          p[t] += w * TANHF(qv + sEnc[b * (T * E) + t * E + e]);
      }
#pragma unroll
      for (int off = 8; off > 0; off >>= 1)
#pragma unroll
        for (int t = 0; t < T; ++t)
          p[t] += __shfl_xor(p[t], off, 16);
      // softmax over t (attn_b2 cancels); every lane holds the full score vector
      float mx = p[0];
#pragma unroll
      for (int t = 1; t < T; ++t) mx = fmaxf(mx, p[t]);
      float s = 0.f;
#pragma unroll
      for (int t = 0; t < T; ++t) { p[t] = __expf(p[t] - mx); s += p[t]; }
      const float inv = RCPF(s);
      float ctx[4];
#pragma unroll
      for (int e4 = 0; e4 < 4; ++e4) {
        const int e = i + e4 * 16;
        float c = 0.f;
#pragma unroll
        for (int t = 0; t < T; ++t)
          c += p[t] * sX[b * (T * E) + t * E + e];
        ctx[e4] = c * inv;
        sCtx[b * E + e] = ctx[e4];
      }
      // fused y_tilde = [context, y_t] @ fc_w^T + fc_b
#pragma unroll
      for (int d = 0; d < DY; ++d) {
        float a = 0.f;
#pragma unroll
        for (int e4 = 0; e4 < 4; ++e4)
          a += ctx[e4] * sFcW[d * 72 + i + e4 * 16];
#pragma unroll
        for (int off = 8; off > 0; off >>= 1)
          a += __shfl_xor(a, off, 16);
        if (i == 0) {
          float acc = a + sFcB[d];
#pragma unroll
          for (int j = 0; j < DY; ++j)
            acc += sY[b * (T * DY) + ts * DY + j] * sFcW[d * 72 + 64 + j];
          sYt[b * 8 + d] = acc;
        }
      }
    }
    __syncthreads();

    // ---- Phase C: gates += y_tilde @ W_ih^T ; LSTM elementwise update ----
    {
      const int b = tid >> 4, i = tid & 15;
      const float y0 = sYt[b * 8 + 0], y1 = sYt[b * 8 + 1], y2 = sYt[b * 8 + 2],
                  y3 = sYt[b * 8 + 3], y4 = sYt[b * 8 + 4];
#pragma unroll
      for (int h4 = 0; h4 < 4; ++h4) {
        const int h = i + h4 * 16;
        float g4[4];
#pragma unroll
        for (int s = 0; s < 4; ++s) {
          const int n = s * 64 + h;
          g4[s] = sGates[b * G4H + n]
                + y0 * sWihT[0 * G4H + n] + y1 * sWihT[1 * G4H + n]
                + y2 * sWihT[2 * G4H + n] + y3 * sWihT[3 * G4H + n]
                + y4 * sWihT[4 * G4H + n];
        }
        const float ii = sigmoidf_fast(g4[0]);
        const float ff = sigmoidf_fast(g4[1]);
        const float oo = sigmoidf_fast(g4[3]);
        const float cn = ff * sC[b * H + h] + ii * TANHF(g4[2]);
        sC[b * H + h] = cn;
        sH[b * H + h] = oo * TANHF(cn);
      }
    }
    __syncthreads();
  }

  // ---------------- Phase 3: out = [ht, context] @ fco_w^T + fco_b ----------------
  if (tid < MB * DY) {
    const int b = tid / DY, d = tid % DY;
    float acc = sFcoB[d];
#pragma unroll
    for (int j = 0; j < H; ++j) acc += sH[b * H + j] * sFcoW[d * 128 + j];
#pragma unroll
    for (int j = 0; j < E; ++j) acc += sCtx[b * E + j] * sFcoW[d * 128 + 64 + j];
    out[(size_t)(b0 + b) * DY + d] = acc;
  }
}

extern "C" void kernel_launch(void* const* d_in, const int* in_sizes, int n_in,
                              void* d_out, int out_size, void* d_ws, size_t ws_size,
                              hipStream_t stream) {
  (void)in_sizes; (void)n_in; (void)out_size;
  const float* x_encoded = (const float*)d_in[0];
  const float* y_history = (const float*)d_in[1];
  const float* ht0       = (const float*)d_in[2];
  const float* ct0       = (const float*)d_in[3];
  const float* attn_w1   = (const float*)d_in[4];
  const float* attn_b1   = (const float*)d_in[5];
  const float* attn_w2   = (const float*)d_in[6];
  // d_in[7] = attn_b2: cancels in softmax
  const float* W_ih      = (const float*)d_in[8];
  const float* W_hh      = (const float*)d_in[9];
  const float* b_ih      = (const float*)d_in[10];
  const float* b_hh      = (const float*)d_in[11];
  const float* fc_w      = (const float*)d_in[12];
  const float* fc_b      = (const float*)d_in[13];
  const float* fco_w     = (const float*)d_in[14];
  const float* fco_b     = (const float*)d_in[15];

  const int B = 65536;
  dim3 grid(B / MB), block(THREADS);

  if (ws_size >= (size_t)(64 * G4H) * sizeof(float)) {
    float* WhhT = (float*)d_ws;
    transpose_whh<<<dim3(64), dim3(256), 0, stream>>>(W_hh, WhhT);
    attn_decoder_kernel<false><<<grid, block, 0, stream>>>(
        x_encoded, y_history, ht0, ct0, attn_w1, attn_b1, attn_w2,
        W_ih, W_hh, b_ih, b_hh, fc_w, fc_b, fco_w, fco_b, WhhT, (float*)d_out);
  } else {
    attn_decoder_kernel<true><<<grid, block, 0, stream>>>(
        x_encoded, y_history, ht0, ct0, attn_w1, attn_b1, attn_w2,
        W_ih, W_hh, b_ih, b_hh, fc_w, fc_b, fco_w, fco_b, W_hh, (float*)d_out);
  }
}